// MGNN_16544214024613
// MI455X (gfx1250) — compile-verified
//
#include <hip/hip_runtime.h>
#include <hip/hip_bf16.h>

#define D_DIM 128
#define BM 64
#define NPATH 3
#define LDS_PITCH 130   // 128 dwords + 2 pad dwords per row (TDM pad feature)

typedef __attribute__((ext_vector_type(2))) float v2f;
typedef __attribute__((ext_vector_type(8))) float v8f;
typedef __attribute__((ext_vector_type(4))) unsigned int v4u;
typedef __attribute__((ext_vector_type(8))) int v8i;
typedef __attribute__((ext_vector_type(4))) int v4i;

// ---------------------------------------------------------------------------
// Tensor Data Mover: async 2D tile load Global -> LDS with row padding.
// Descriptor packing per CDNA5 ISA ch.8 (D# groups 0/1; groups 2/3 zero for
// 2D tiles).  Row pad: interval 128 DWORDs (code 6), amount 2 DWORDs (code 1)
// -> LDS row pitch 130 floats, which de-conflicts 64-bit DS reads striding
// one row per lane (130 mod 64 == 2 banks apart).
// This toolchain exposes the 6-arg builtin: (v4u, v8i, v4i, v4i, v8i, i32).
// ---------------------------------------------------------------------------
__device__ __forceinline__ void MGNN_tdm_load_2d(
    unsigned lds_off, const void* gaddr, unsigned tensor_d0, unsigned tensor_d1,
    unsigned tile_d0, unsigned tile_d1, unsigned stride0) {
#if __has_builtin(__builtin_amdgcn_tensor_load_to_lds)
    const unsigned long long ga = (unsigned long long)gaddr;
    v4u g0;
    g0.x = 1u;                                         // count=1 (valid D#)
    g0.y = lds_off;                                    // lds_addr (bytes)
    g0.z = (unsigned)(ga & 0xFFFFFFFFu);               // global_addr[31:0]
    g0.w = (unsigned)((ga >> 32) & 0x01FFFFFFu) | (2u << 30);  // addr[56:32], type=2
    v8i g1;
    g1[0] = (int)((2u << 16)        // data_size = 4 bytes
                | (1u << 20)        // pad_enable
                | (6u << 22)        // pad_interval: 128 DWORDs
                | (1u << 25));      // pad_amount: 2 DWORDs
    g1[1] = (int)((tensor_d0 & 0xFFFFu) << 16);                        // dim0[15:0]
    g1[2] = (int)(((tensor_d0 >> 16) & 0xFFFFu) | ((tensor_d1 & 0xFFFFu) << 16));
    g1[3] = (int)(((tensor_d1 >> 16) & 0xFFFFu) | ((tile_d0 & 0xFFFFu) << 16));
    g1[4] = (int)(tile_d1 & 0xFFFFu);                                  // tile_dim1, tile_dim2=0
    g1[5] = (int)stride0;                                              // dim0_stride[31:0]
    g1[6] = 0;                                                         // stride hi / dim1_stride
    g1[7] = 0;
    v4i g2 = {0, 0, 0, 0};
    v4i g3 = {0, 0, 0, 0};
    v8i g4 = {0, 0, 0, 0, 0, 0, 0, 0};
    __builtin_amdgcn_tensor_load_to_lds(g0, g1, g2, g3, g4, 0);
#else
    (void)lds_off; (void)gaddr; (void)tensor_d0; (void)tensor_d1;
    (void)tile_d0; (void)tile_d1; (void)stride0;
#endif
}

// ---------------------------------------------------------------------------
// Zero-fill helper (agg / scores must be zeroed every call; harness poisons ws)
// ---------------------------------------------------------------------------
__global__ void MGNN_zero_kernel(float* __restrict__ p, size_t n) {
    size_t i = (size_t)blockIdx.x * blockDim.x + threadIdx.x;
    if (i < n) p[i] = 0.0f;
}

// ---------------------------------------------------------------------------
// C[M x 128] = A[M x 128] @ W[128 x 128]^T   (C[m,n] = sum_k A[m,k] * W[n,k])
// 256 threads = 8 wave32; block computes a 64x128 tile of C.
// W tile (128x128) and A tile (64x128) staged in LDS by the TDM (wave 0
// issues tensor_load_to_lds, waits s_wait_tensorcnt, then workgroup barrier).
// Both LDS tiles use pitch-130 rows so A/B fragment ds_load_b64 reads are
// bank-conflict-free and land directly in even-aligned WMMA operand pairs.
// Store epilogue: uniform full-tile fast path (no per-element exec masking);
// guarded path only for the final partial row block.
// ---------------------------------------------------------------------------
__global__ __launch_bounds__(256) void MGNN_gemm_xWT(
    const float* __restrict__ A, const float* __restrict__ W,
    float* __restrict__ C, int M) {
    __shared__ float lds[(D_DIM + BM) * LDS_PITCH];
    float* Wlds = lds;                          // [128][130]: Wlds[n][k] = W[n*128+k]
    float* Alds = lds + D_DIM * LDS_PITCH;      // [ 64][130]: Alds[r][k] = A[(row0+r)*128+k]
    const int tid = threadIdx.x;
    const int blockRow0 = blockIdx.x * BM;

#if __has_builtin(__builtin_amdgcn_tensor_load_to_lds)
    if (tid < 32) {
        MGNN_tdm_load_2d((unsigned)(uintptr_t)Wlds, W,
                         D_DIM, D_DIM, D_DIM, D_DIM, D_DIM);
        MGNN_tdm_load_2d((unsigned)(uintptr_t)Alds, A + (size_t)blockRow0 * D_DIM,
                         D_DIM, (unsigned)(M - blockRow0), D_DIM, BM, D_DIM);
        __builtin_amdgcn_s_wait_tensorcnt(0);
    }
#else
    for (int idx = tid; idx < D_DIM * D_DIM; idx += 256) {
        const int n = idx >> 7, k = idx & 127;
        Wlds[n * LDS_PITCH + k] = W[idx];
    }
    for (int idx = tid; idx < BM * D_DIM; idx += 256) {
        const int r = idx >> 7, k = idx & 127;
        const int gr = blockRow0 + r;
        Alds[r * LDS_PITCH + k] = (gr < M) ? A[(size_t)gr * D_DIM + k] : 0.0f;
    }
#endif
    __syncthreads();

    const int wave = tid >> 5, lane = tid & 31;
    const int rb = wave >> 1, ch = wave & 1;    // row band / column half
    const int rowBase = blockRow0 + rb * 16;
    const int colBase = ch * 64;
    const int m16 = lane & 15;                  // row (A) / col (B) within tile
    const int khalf = lane >> 4;                // lanes 0-15: K={0,1}; 16-31: K={2,3}

    v8f acc[4] = {};
    const float* Ap = Alds + (rb * 16 + m16) * LDS_PITCH + khalf * 2;
    const float* Bp = Wlds + (colBase + m16) * LDS_PITCH + khalf * 2;

    for (int k0 = 0; k0 < D_DIM; k0 += 4) {
        const v2f a = *(const v2f*)(Ap + k0);   // ds_load_b64, conflict-free
#pragma unroll
        for (int t = 0; t < 4; ++t) {
            const v2f b = *(const v2f*)(Bp + t * 16 * LDS_PITCH + k0);
            acc[t] = __builtin_amdgcn_wmma_f32_16x16x4_f32(
                false, a, false, b, (short)0, acc[t], false, false);
        }
    }

    const int srow = rowBase + khalf * 8;       // this lane's first output row
    if (blockRow0 + BM <= M) {                  // uniform: whole 64-row tile valid
#pragma unroll
        for (int t = 0; t < 4; ++t) {
            float* cp = C + (size_t)srow * D_DIM + colBase + t * 16 + m16;
#pragma unroll
            for (int r = 0; r < 8; ++r) cp[(size_t)r * D_DIM] = acc[t][r];
        }
    } else {                                    // final partial block only
#pragma unroll
        for (int t = 0; t < 4; ++t) {
            const int col = colBase + t * 16 + m16;
#pragma unroll
            for (int r = 0; r < 8; ++r) {
                const int row = srow + r;
                if (row < M) C[(size_t)row * D_DIM + col] = acc[t][r];
            }
        }
    }
}

// ---------------------------------------------------------------------------
// SpMM scatter: agg[row] += val * seq_fts[col].  One wave32 per edge,
// lane*4 floats each (float4 gather + 4 global f32 atomics; seq_fts and agg
// are 25.6 MB each so the whole scatter stays L2-resident on 192 MB L2).
// ---------------------------------------------------------------------------
__global__ void MGNN_spmm_kernel(
    const float* __restrict__ seq, const int* __restrict__ rows,
    const int* __restrict__ cols, const float* __restrict__ vals,
    float* __restrict__ agg, int E) {
    const int gid = blockIdx.x * blockDim.x + threadIdx.x;
    const int e = gid >> 5, lane = gid & 31;
    if (e >= E) return;
    const int r = rows[e], c = cols[e];
    const float v = vals[e];
    const float4 x = *(const float4*)(seq + (size_t)c * D_DIM + lane * 4);
    float* dst = agg + (size_t)r * D_DIM + lane * 4;
    unsafeAtomicAdd(dst + 0, v * x.x);
    unsafeAtomicAdd(dst + 1, v * x.y);
    unsafeAtomicAdd(dst + 2, v * x.z);
    unsafeAtomicAdd(dst + 3, v * x.w);
}

// ---------------------------------------------------------------------------
// FiLM affine + bias + residual + PReLU.  emb = PReLU(g*agg + b + fb + seq)
// ---------------------------------------------------------------------------
__global__ void MGNN_film_kernel(
    const float* __restrict__ seq, const float* __restrict__ agg,
    const int* __restrict__ ntype, const float* __restrict__ Wg,
    const float* __restrict__ bg, const float* __restrict__ Wb,
    const float* __restrict__ bb, const float* __restrict__ fbias,
    const float* __restrict__ prelu_a, int p, float* __restrict__ emb, int Nn) {
    const int gid = blockIdx.x * blockDim.x + threadIdx.x;
    const int n = gid >> 5, j = gid & 31;
    if (n >= Nn) return;
    const int t = ntype[n];
    const float a = prelu_a[p];
    const int d0 = j * 4;
    const size_t off = (size_t)n * D_DIM + d0;
#pragma unroll
    for (int q = 0; q < 4; ++q) {
        const int d = d0 + q;
        const float g = Wg[d * 2 + t] + bg[d];
        const float be = Wb[d * 2 + t] + bb[d];
        float o = g * agg[off + q] + be + fbias[d] + seq[off + q];
        emb[off + q] = (o >= 0.0f) ? o : a * o;
    }
}

// ---------------------------------------------------------------------------
// Attention score: scores[n] += w2 . tanh(att_W1 @ z[n] + b1)
// Same TDM + WMMA tiling, with the 128-wide reduction fused into the
// epilogue (shfl_xor across the 16 column lanes, atomic combine across the
// two column-half waves).
// ---------------------------------------------------------------------------
__global__ __launch_bounds__(256) void MGNN_attn_score(
    const float* __restrict__ Z, const float* __restrict__ W1,
    const float* __restrict__ b1, const float* __restrict__ w2,
    float* __restrict__ scores, int M) {
    __shared__ float lds[(D_DIM + BM) * LDS_PITCH];
    float* Wlds = lds;
    float* Alds = lds + D_DIM * LDS_PITCH;
    const int tid = threadIdx.x;
    const int blockRow0 = blockIdx.x * BM;

#if __has_builtin(__builtin_amdgcn_tensor_load_to_lds)
    if (tid < 32) {
        MGNN_tdm_load_2d((unsigned)(uintptr_t)Wlds, W1,
                         D_DIM, D_DIM, D_DIM, D_DIM, D_DIM);
        MGNN_tdm_load_2d((unsigned)(uintptr_t)Alds, Z + (size_t)blockRow0 * D_DIM,
                         D_DIM, (unsigned)(M - blockRow0), D_DIM, BM, D_DIM);
        __builtin_amdgcn_s_wait_tensorcnt(0);
    }
#else
    for (int idx = tid; idx < D_DIM * D_DIM; idx += 256) {
        const int n = idx >> 7, k = idx & 127;
        Wlds[n * LDS_PITCH + k] = W1[idx];
    }
    for (int idx = tid; idx < BM * D_DIM; idx += 256) {
        const int r = idx >> 7, k = idx & 127;
        const int gr = blockRow0 + r;
        Alds[r * LDS_PITCH + k] = (gr < M) ? Z[(size_t)gr * D_DIM + k] : 0.0f;
    }
#endif
    __syncthreads();

    const int wave = tid >> 5, lane = tid & 31;
    const int rb = wave >> 1, ch = wave & 1;
    const int rowBase = blockRow0 + rb * 16;
    const int colBase = ch * 64;
    const int m16 = lane & 15;
    const int khalf = lane >> 4;

    v8f acc[4] = {};
    const float* Ap = Alds + (rb * 16 + m16) * LDS_PITCH + khalf * 2;
    const float* Bp = Wlds + (colBase + m16) * LDS_PITCH + khalf * 2;

    for (int k0 = 0; k0 < D_DIM; k0 += 4) {
        const v2f a = *(const v2f*)(Ap + k0);
#pragma unroll
        for (int t = 0; t < 4; ++t) {
            const v2f b = *(const v2f*)(Bp + t * 16 * LDS_PITCH + k0);
            acc[t] = __builtin_amdgcn_wmma_f32_16x16x4_f32(
                false, a, false, b, (short)0, acc[t], false, false);
        }
    }

    // epilogue: s[r] = sum over this wave's 64 columns of tanh(acc+b1)*w2
    float s[8] = {0.f, 0.f, 0.f, 0.f, 0.f, 0.f, 0.f, 0.f};
#pragma unroll
    for (int t = 0; t < 4; ++t) {
        const int col = colBase + t * 16 + m16;
        const float b1v = b1[col];
        const float w2v = w2[col];
#pragma unroll
        for (int r = 0; r < 8; ++r) s[r] += tanhf(acc[t][r] + b1v) * w2v;
    }
    const bool fullTile = (blockRow0 + BM <= M);   // uniform
#pragma unroll
    for (int r = 0; r < 8; ++r) {
        float v = s[r];
        v += __shfl_xor(v, 1);
        v += __shfl_xor(v, 2);
        v += __shfl_xor(v, 4);
        v += __shfl_xor(v, 8);   // reduces within each 16-lane half
        const int row = rowBase + khalf * 8 + r;
        if (m16 == 0 && (fullTile || row < M)) unsafeAtomicAdd(&scores[row], v);
    }
}

// ---------------------------------------------------------------------------
// Softmax over P metapaths + weighted combine + residual h.
// ---------------------------------------------------------------------------
__global__ void MGNN_final_kernel(
    const float* __restrict__ emb, const float* __restrict__ scores,
    const float* __restrict__ h, float* __restrict__ out, int Nn) {
    const int gid = blockIdx.x * blockDim.x + threadIdx.x;
    const int n = gid >> 5, j = gid & 31;
    if (n >= Nn) return;
    const size_t NND = (size_t)Nn * D_DIM;
    const float s0 = scores[n];
    const float s1 = scores[Nn + n];
    const float s2 = scores[2 * (size_t)Nn + n];
    const float m = fmaxf(s0, fmaxf(s1, s2));
    const float e0 = __expf(s0 - m), e1 = __expf(s1 - m), e2 = __expf(s2 - m);
    const float inv = 1.0f / (e0 + e1 + e2);
    const float w0 = e0 * inv, w1 = e1 * inv, w2 = e2 * inv;
    const size_t off = (size_t)n * D_DIM + j * 4;
    const float4 z0 = *(const float4*)(emb + off);
    const float4 z1 = *(const float4*)(emb + NND + off);
    const float4 z2 = *(const float4*)(emb + 2 * NND + off);
    const float4 hh = *(const float4*)(h + off);
    float4 o;
    o.x = w0 * z0.x + w1 * z1.x + w2 * z2.x + hh.x;
    o.y = w0 * z0.y + w1 * z1.y + w2 * z2.y + hh.y;
    o.z = w0 * z0.z + w1 * z1.z + w2 * z2.z + hh.z;
    o.w = w0 * z0.w + w1 * z1.w + w2 * z2.w + hh.w;
    *(float4*)(out + off) = o;
}

// ---------------------------------------------------------------------------
extern "C" void kernel_launch(void* const* d_in, const int* in_sizes, int n_in,
                              void* d_out, int out_size, void* d_ws, size_t ws_size,
                              hipStream_t stream) {
    const float* h        = (const float*)d_in[0];
    const int*   erows    = (const int*)d_in[1];
    const int*   ecols    = (const int*)d_in[2];
    const float* evals    = (const float*)d_in[3];
    const int*   ntype    = (const int*)d_in[4];
    const float* W_fc     = (const float*)d_in[5];
    const float* prelu_a  = (const float*)d_in[6];
    const float* Wg       = (const float*)d_in[7];
    const float* bg       = (const float*)d_in[8];
    const float* Wb       = (const float*)d_in[9];
    const float* bb       = (const float*)d_in[10];
    const float* fbias    = (const float*)d_in[11];
    const float* attW1    = (const float*)d_in[12];
    const float* attb1    = (const float*)d_in[13];
    const float* attw2    = (const float*)d_in[14];
    float* out = (float*)d_out;

    const int N = in_sizes[0] / D_DIM;
    const int E = in_sizes[1] / NPATH;

    // Workspace layout (floats): seq_fts[N*D] | agg[N*D] | emb[P*N*D] | scores[P*N]
    float* seq    = (float*)d_ws;
    float* agg    = seq + (size_t)N * D_DIM;
    float* emb    = agg + (size_t)N * D_DIM;
    float* scores = emb + (size_t)NPATH * N * D_DIM;

    const dim3 blk(256);
    const int gemmGrid = (N + BM - 1) / BM;
    const int elemGrid = (N * 32 + 255) / 256;          // one lane per 4 floats
    const size_t ndCount = (size_t)N * D_DIM;

    MGNN_zero_kernel<<<(NPATH * N + 255) / 256, blk, 0, stream>>>(scores,
                                                                  (size_t)NPATH * N);

    for (int p = 0; p < NPATH; ++p) {
        MGNN_gemm_xWT<<<gemmGrid, blk, 0, stream>>>(
            h, W_fc + (size_t)p * D_DIM * D_DIM, seq, N);
        MGNN_zero_kernel<<<(int)((ndCount + 255) / 256), blk, 0, stream>>>(agg, ndCount);
        MGNN_spmm_kernel<<<(E * 32 + 255) / 256, blk, 0, stream>>>(
            seq, erows + (size_t)p * E, ecols + (size_t)p * E,
            evals + (size_t)p * E, agg, E);
        MGNN_film_kernel<<<elemGrid, blk, 0, stream>>>(
            seq, agg, ntype, Wg + (size_t)p * D_DIM * 2, bg + (size_t)p * D_DIM,
            Wb + (size_t)p * D_DIM * 2, bb + (size_t)p * D_DIM,
            fbias + (size_t)p * D_DIM, prelu_a, p,
            emb + (size_t)p * N * D_DIM, N);
        MGNN_attn_score<<<gemmGrid, blk, 0, stream>>>(
            emb + (size_t)p * N * D_DIM, attW1, attb1, attw2, scores + (size_t)p * N, N);
    }

    MGNN_final_kernel<<<elemGrid, blk, 0, stream>>>(emb, scores, h, out, N);
}